// primary_capsule_19688130084892
// MI455X (gfx1250) — compile-verified
//
#include <hip/hip_runtime.h>
#include <math.h>

#define BATCH 128
#define OUTC  64
#define JDIM  32
#define KDIM  16
#define DDIM  1024

typedef __attribute__((ext_vector_type(2))) float v2f;
typedef __attribute__((ext_vector_type(8))) float v8f;

// ---------------------------------------------------------------------------
// Kernel 1: Xt[d][j][b] = X[b][j][d]
// float4 (b128) reads along d; transpose via LDS; b-contiguous b32 writes.
// Tile: 32 b x 128 d for one j.  grid: (D/128, B/32, J), block: (32, 8)
// ---------------------------------------------------------------------------
__global__ void __launch_bounds__(256)
transpose_x_kernel(const float* __restrict__ X, float* __restrict__ Xt) {
    __shared__ float tile[32 * 133];          // [b_local][d_local], stride 133
    const int j  = blockIdx.z;
    const int d0 = blockIdx.x * 128;
    const int b0 = blockIdx.y * 32;
    const int tx = threadIdx.x;               // 0..31
    const int ty = threadIdx.y;               // 0..7
    #pragma unroll
    for (int r = ty; r < 32; r += 8) {
        const float4 v = *(const float4*)(
            X + ((size_t)(b0 + r) * JDIM + j) * DDIM + d0 + tx * 4);
        float* t = &tile[r * 133 + tx * 4];
        t[0] = v.x; t[1] = v.y; t[2] = v.z; t[3] = v.w;
    }
    __syncthreads();
    #pragma unroll
    for (int dl = ty; dl < 128; dl += 8)      // lanes cover 32 b -> 128B stores
        Xt[((size_t)(d0 + dl) * JDIM + j) * BATCH + b0 + tx] = tile[tx * 133 + dl];
}

// ---------------------------------------------------------------------------
// Kernel 2: Wr[d][j][i] = sum_k W[i][j][k][d]
// The single pass over the 134MB weight tensor: float4 (b128) reads along d,
// k-reduction in registers, transpose via LDS, i-contiguous writes.
// Tile: 32 i x 128 d for one j.  grid: (D/128, I/32, J), block: (32, 8)
// ---------------------------------------------------------------------------
__global__ void __launch_bounds__(256)
reduce_w_kernel(const float* __restrict__ W, float* __restrict__ Wr) {
    __shared__ float tile[32 * 133];          // [i_local][d_local], stride 133
    const int j  = blockIdx.z;
    const int d0 = blockIdx.x * 128;
    const int i0 = blockIdx.y * 32;
    const int tx = threadIdx.x;
    const int ty = threadIdx.y;
    #pragma unroll
    for (int r = ty; r < 32; r += 8) {
        const float* base =
            W + ((size_t)((i0 + r) * JDIM + j) * KDIM) * DDIM + d0 + tx * 4;
        float sx = 0.f, sy = 0.f, sz = 0.f, sw = 0.f;
        #pragma unroll
        for (int k = 0; k < KDIM; ++k) {
            const float4 v = *(const float4*)(base + (size_t)k * DDIM);
            sx += v.x; sy += v.y; sz += v.z; sw += v.w;
        }
        float* t = &tile[r * 133 + tx * 4];
        t[0] = sx; t[1] = sy; t[2] = sz; t[3] = sw;
    }
    __syncthreads();
    #pragma unroll
    for (int dl = ty; dl < 128; dl += 8)      // lanes cover 32 i -> 128B stores
        Wr[((size_t)(d0 + dl) * JDIM + j) * OUTC + i0 + tx] = tile[tx * 133 + dl];
}

// ---------------------------------------------------------------------------
// Kernel 3: per-d GEMM  S_d[16b x 16i] = X_d[16x32] * Wr_d[32x16]
// via V_WMMA_F32_16X16X4_F32 (8 steps over j=32). Block tile 16b x 16i x 32d,
// 8 waves, wave w owns d-slots {4w..4w+3}. Results transposed through padded
// LDS so the [B][I][D] stores are lane-contiguous in d (128B coalesced).
// grid: (D/32, B/16, I/16), block: 256
// ---------------------------------------------------------------------------
#define ROWS   20            // LDS row stride (16 cols + pad): halves -> disjoint banks
#define DLS    321           // LDS d-slot stride: lane*321 % 64 == lane (conflict-free)
__global__ void __launch_bounds__(256)
caps_wmma_kernel(const float* __restrict__ Xt, const float* __restrict__ Wr,
                 float* __restrict__ S) {
    __shared__ float stile[32 * DLS];   // 41,088 B
    const int d0   = blockIdx.x * 32;
    const int b0   = blockIdx.y * 16;
    const int i0   = blockIdx.z * 16;
    const int tid  = threadIdx.x;
    const int wave = tid >> 5;
    const int lane = tid & 31;
    const int half = lane >> 4;         // 0: lanes 0-15, 1: lanes 16-31
    const int l16  = lane & 15;

    #pragma unroll
    for (int dd = 0; dd < 4; ++dd) {
        const int dl = wave * 4 + dd;
        const int d  = d0 + dl;
        const float* xrow = Xt + (size_t)d * JDIM * BATCH + b0 + l16;
        const float* wrow = Wr + (size_t)d * JDIM * OUTC  + i0 + l16;
        v8f acc = {};
        #pragma unroll
        for (int c = 0; c < 8; ++c) {
            const int jA = c * 4 + half * 2;
            v2f a, b;
            a.x = xrow[(size_t)jA * BATCH];
            a.y = xrow[(size_t)(jA + 1) * BATCH];
            b.x = wrow[(size_t)jA * OUTC];
            b.y = wrow[(size_t)(jA + 1) * OUTC];
            acc = __builtin_amdgcn_wmma_f32_16x16x4_f32(
                /*neg_a=*/false, a, /*neg_b=*/false, b,
                /*c_mod=*/(short)0, acc, /*reuse_a=*/false, /*reuse_b=*/false);
        }
        // C/D layout: VGPR r -> (M = r + 8*half, N = l16)
        #pragma unroll
        for (int r = 0; r < 8; ++r)
            stile[dl * DLS + (r + half * 8) * ROWS + l16] = acc[r];
    }
    __syncthreads();
    // Each wave writes 32 (b,i) pairs; lanes cover the 32 d-slots.
    #pragma unroll
    for (int idx = 0; idx < 32; ++idx) {
        const int p  = wave * 32 + idx;
        const int pb = p >> 4;
        const int pi = p & 15;
        const float v = stile[lane * DLS + pb * ROWS + pi];
        S[((size_t)(b0 + pb) * OUTC + (i0 + pi)) * DDIM + d0 + lane] = v;
    }
}

// ---------------------------------------------------------------------------
// Kernel 4: fused squash — one block per (b,i) pair. Row values stay in
// registers across the fixed-order LDS tree reduction (deterministic), then
// the scale is applied in place. Single pass over d_out.
// scale = ss / ((1+ss) * (sqrt(ss)+1e-8))
// grid: B*I, block: 256
// ---------------------------------------------------------------------------
__global__ void __launch_bounds__(256)
squash_kernel(float* __restrict__ S) {
    __shared__ float red[256];
    const size_t pair = blockIdx.x;
    float* row = S + pair * DDIM;
    float v[4];
    float s = 0.f;
    #pragma unroll
    for (int u = 0; u < 4; ++u) {
        v[u] = row[threadIdx.x + u * 256];
        s += v[u] * v[u];
    }
    red[threadIdx.x] = s;
    __syncthreads();
    #pragma unroll
    for (int off = 128; off > 0; off >>= 1) {
        if (threadIdx.x < off) red[threadIdx.x] += red[threadIdx.x + off];
        __syncthreads();
    }
    const float ss = red[0];
    const float sc = ss / ((1.f + ss) * (sqrtf(ss) + 1e-8f));
    #pragma unroll
    for (int u = 0; u < 4; ++u) row[threadIdx.x + u * 256] = v[u] * sc;
}

// ---------------------------------------------------------------------------
extern "C" void kernel_launch(void* const* d_in, const int* in_sizes, int n_in,
                              void* d_out, int out_size, void* d_ws, size_t ws_size,
                              hipStream_t stream) {
    const float* X = (const float*)d_in[0];   // [B, J, D]
    const float* W = (const float*)d_in[1];   // [I, J, K, D]
    float* out = (float*)d_out;               // [B, I, D] (also holds s)

    float* Xt = (float*)d_ws;                         // D*J*B = 16 MB
    float* Wr = Xt + (size_t)DDIM * JDIM * BATCH;     // D*J*I =  8 MB

    transpose_x_kernel<<<dim3(DDIM / 128, BATCH / 32, JDIM), dim3(32, 8), 0, stream>>>(X, Xt);
    reduce_w_kernel  <<<dim3(DDIM / 128, OUTC / 32, JDIM),  dim3(32, 8), 0, stream>>>(W, Wr);
    caps_wmma_kernel <<<dim3(DDIM / 32, BATCH / 16, OUTC / 16), 256, 0, stream>>>(Xt, Wr, out);
    squash_kernel    <<<dim3(BATCH * OUTC), 256, 0, stream>>>(out);
}